// LossNet_63410897158253
// MI455X (gfx1250) — compile-verified
//
#include <hip/hip_runtime.h>

typedef __attribute__((ext_vector_type(2))) float v2f;
typedef __attribute__((ext_vector_type(8))) float v8f;

#define BLOCK 256
// B=8, S=1024, V=32000 ; valid elements = 8*1023 = 8184 -> 32 blocks of 256
#define NBLOCKS 32

__global__ __launch_bounds__(BLOCK) void loss_gather_partial(
    const int* __restrict__ cap_index,
    const float* __restrict__ pred,
    float* __restrict__ partials,
    int total)
{
    __shared__ float smem[BLOCK];
    const int tid = threadIdx.x;
    const int gid = blockIdx.x * BLOCK + tid;

    float v = 0.0f;
    if (gid < total) {
        const int b   = gid / 1023;
        const int s   = gid - b * 1023;                 // s in [0, 1022]
        const int idx = cap_index[b * 1024 + s + 1];    // tgt = cap_index[:,1:]
        const size_t row = (size_t)(b * 1024 + s);      // pred[b, s, :]
        const float p = pred[row * 32000u + (size_t)idx];
        const float t = 1.0f - (float)s * (1.0f / 1024.0f);  // exact in f32
        v = -(t * t) * logf(p);
    }
    smem[tid] = v;
    __syncthreads();

    // Wave 0 reduces all 256 values via the f32 WMMA pipe.
    // A (16x4 f32) holds 64 loss terms; B = ones(4x16) => D[m,n] = row-sum of A.
    // Chain 4 accumulating WMMAs to cover all 256 values, then total =
    // sum over column 0 = lane0(Sum VGPR0..7, M=0..7) + lane16(Sum VGPR0..7, M=8..15).
    if (tid < 32) {                 // wave 0: EXEC all-1s inside this branch
        v8f acc = {};
        v2f ones;
        ones.x = 1.0f;
        ones.y = 1.0f;
#pragma unroll
        for (int c = 0; c < 4; ++c) {
            v2f a;
            a.x = smem[c * 64 + tid];        // K=0 (lanes 0-15), K=2 (lanes 16-31)
            a.y = smem[c * 64 + 32 + tid];   // K=1 (lanes 0-15), K=3 (lanes 16-31)
            // 8 args: (neg_a, A, neg_b, B, c_mod, C, reuse_a, reuse_b)
            acc = __builtin_amdgcn_wmma_f32_16x16x4_f32(
                false, a, false, ones, (short)0, acc, false, false);
        }
        float col = acc[0] + acc[1] + acc[2] + acc[3] +
                    acc[4] + acc[5] + acc[6] + acc[7];
        float hi = __shfl(col, 16, 32);      // lane16 holds M=8..15 of column 0
        if (tid == 0) partials[blockIdx.x] = col + hi;
    }
}

__global__ __launch_bounds__(32) void loss_final_reduce(
    const float* __restrict__ partials,
    float* __restrict__ out,
    int n)
{
    const int lane = threadIdx.x;
    float v = (lane < n) ? partials[lane] : 0.0f;
#pragma unroll
    for (int off = 16; off > 0; off >>= 1)
        v += __shfl_down(v, off, 32);        // fixed butterfly: deterministic
    if (lane == 0) out[0] = v;
}

extern "C" void kernel_launch(void* const* d_in, const int* in_sizes, int n_in,
                              void* d_out, int out_size, void* d_ws, size_t ws_size,
                              hipStream_t stream)
{
    const int*   cap_index = (const int*)d_in[0];   // (8, 1024) int32
    const float* pred      = (const float*)d_in[1]; // (8, 1024, 32000) f32
    float* out      = (float*)d_out;                // scalar loss
    float* partials = (float*)d_ws;                 // 32 floats of scratch

    const int total = 8 * 1023;                     // 8184 gathered terms

    loss_gather_partial<<<NBLOCKS, BLOCK, 0, stream>>>(cap_index, pred, partials, total);
    loss_final_reduce<<<1, 32, 0, stream>>>(partials, out, NBLOCKS);
}